// Ginformer_32985348833839
// MI455X (gfx1250) — compile-verified
//
#include <hip/hip_runtime.h>

#define NN 20000
#define EE 320000
#define CC 128
#define HH 8
#define DHH 16
#define C3 384
#define CUTOFF_R 5.0f
#define LN_EPSF 1e-5f

typedef __attribute__((ext_vector_type(2))) float v2f;
typedef __attribute__((ext_vector_type(8))) float v8f;
typedef __attribute__((ext_vector_type(4))) int v4i;

#if __has_builtin(__builtin_amdgcn_global_load_async_to_lds_b128) && \
    __has_builtin(__builtin_amdgcn_s_wait_asynccnt)
#define HAVE_ASYNC_LDS 1
typedef __attribute__((address_space(1))) v4i gv4i_t;
typedef __attribute__((address_space(3))) v4i lv4i_t;
__device__ __forceinline__ void async_copy16(const void* g, void* l) {
    __builtin_amdgcn_global_load_async_to_lds_b128(
        (gv4i_t*)g, (lv4i_t*)l, /*offset=*/0, /*cpol=*/0);
}
__device__ __forceinline__ void async_wait0() {
    __builtin_amdgcn_s_wait_asynccnt(0);
}
#else
#define HAVE_ASYNC_LDS 0
#endif

__device__ __forceinline__ float silu_f(float x) { return x / (1.0f + __expf(-x)); }

// monotone float <-> uint mapping for atomicMax on signed floats
__device__ __forceinline__ unsigned fenc(float f) {
    unsigned u = __float_as_uint(f);
    return (u & 0x80000000u) ? ~u : (u | 0x80000000u);
}
__device__ __forceinline__ float fdec(unsigned u) {
    unsigned b = (u & 0x80000000u) ? (u & 0x7FFFFFFFu) : ~u;
    return __uint_as_float(b);
}

__device__ __forceinline__ v8f wmma_f32(v2f a, v2f b, v8f c) {
    return __builtin_amdgcn_wmma_f32_16x16x4_f32(false, a, false, b, (short)0, c,
                                                 false, false);
}

// stage a 16 x 128 float tile (row stride in LDS = CC+4) from global into LDS.
// src row i comes from g + (size_t)(row0+i)*CC. 256 threads, 2 x b128 each.
template <int NT>
__device__ __forceinline__ void stage_tile(const float* __restrict__ g,
                                           size_t row0, float (*T)[CC + 4],
                                           int t) {
#pragma unroll
    for (int i = 0; i < 512 / NT; i++) {
        int f = t + i * NT;          // float4 index in [0,512)
        int rr = f >> 5;             // 32 float4 per row
        int c4 = (f & 31) * 4;
#if HAVE_ASYNC_LDS
        async_copy16(&g[(row0 + rr) * CC + c4], &T[rr][c4]);
#else
        *(float4*)&T[rr][c4] = *(const float4*)&g[(row0 + rr) * CC + c4];
#endif
    }
}

__device__ __forceinline__ void stage_fence() {
#if HAVE_ASYNC_LDS
    async_wait0();
#endif
    __syncthreads();
}

// ---------------------------------------------------------------- K0: init
__global__ __launch_bounds__(256) void k_init(const float* __restrict__ r,
                                              const float* __restrict__ vin,
                                              float* __restrict__ out_v,
                                              float* __restrict__ out_r,
                                              unsigned* __restrict__ mmax,
                                              float* __restrict__ zsum) {
    const int EC4 = EE * CC / 4;
    const int VC4 = NN * 3 * CC / 4;
    const int MH = NN * HH;
    int i = blockIdx.x * 256 + threadIdx.x;
    if (i < EC4) ((float4*)out_r)[i] = ((const float4*)r)[i];
    if (i < VC4) ((float4*)out_v)[i] = ((const float4*)vin)[i];
    if (i < MH) { mmax[i] = 0x007FFFFFu; /* enc(-inf) */ zsum[i] = 0.0f; }
}

// ---------------------------------------- K1: LayerNorm + 4x (C x C) GEMMs
__global__ __launch_bounds__(256) void k_node1(
    const float* __restrict__ s, const float* __restrict__ ln_g,
    const float* __restrict__ ln_b, const float* __restrict__ Wq,
    const float* __restrict__ bq, const float* __restrict__ Wk,
    const float* __restrict__ bk, const float* __restrict__ Wg1,
    const float* __restrict__ bg1, const float* __restrict__ Wv1,
    const float* __restrict__ bv1, float* __restrict__ out_s,
    float* __restrict__ q, float* __restrict__ k, float* __restrict__ h1,
    float* __restrict__ hv) {
    __shared__ float sT[16][CC + 4];
    __shared__ float mu[16], rsd[16];
    const int t = threadIdx.x;
    const int n0 = blockIdx.x * 16;
    stage_tile<256>(s, (size_t)n0, sT, t);
    stage_fence();
    if (t < 16) {
        float sum = 0.f, sq = 0.f;
#pragma unroll 8
        for (int c = 0; c < CC; c++) {
            float x = sT[t][c];
            sum += x; sq += x * x;
        }
        float m = sum * (1.0f / CC);
        float var = sq * (1.0f / CC) - m * m;
        mu[t] = m;
        rsd[t] = rsqrtf(var + LN_EPSF);
    }
    __syncthreads();
#pragma unroll
    for (int i = 0; i < 8; i++) {
        int e = t + i * 256;
        int rr = e >> 7, col = e & 127;
        float x = (sT[rr][col] - mu[rr]) * rsd[rr] * ln_g[col] + ln_b[col];
        sT[rr][col] = x;
        out_s[(size_t)(n0 + rr) * CC + col] = x;  // s_out = s_ln (+ su later)
    }
    __syncthreads();
    const int wave = t >> 5, lane = t & 31;
    const int l16 = lane & 15, hi = lane >> 4;
    const int nb = wave * 16 + l16;
    const float* Ws[4] = {Wq, Wk, Wg1, Wv1};
    const float* Bs[4] = {bq, bk, bg1, bv1};
    float* Ds[4] = {q, k, h1, hv};
    for (int m = 0; m < 4; m++) {
        const float* W = Ws[m];
        v8f acc = {0, 0, 0, 0, 0, 0, 0, 0};
#pragma unroll 8
        for (int k0 = 0; k0 < CC; k0 += 4) {
            int ka = k0 + hi * 2;
            v2f a, b;
            a.x = sT[l16][ka]; a.y = sT[l16][ka + 1];
            b.x = W[(size_t)ka * CC + nb]; b.y = W[(size_t)(ka + 1) * CC + nb];
            acc = wmma_f32(a, b, acc);
        }
        float bias = Bs[m][nb];
        bool act = (m >= 2);
        float* D = Ds[m];
#pragma unroll
        for (int vv = 0; vv < 8; vv++) {
            int rr = vv + hi * 8;
            float dv = acc[vv] + bias;
            if (act) dv = silu_f(dv);
            D[(size_t)(n0 + rr) * CC + nb] = dv;
        }
    }
}

// ------------------------------------------- K2: 2x (C x 3C) GEMMs (x3,val)
__global__ __launch_bounds__(256) void k_node2(
    const float* __restrict__ h1, const float* __restrict__ hv,
    const float* __restrict__ Wg2, const float* __restrict__ bg2,
    const float* __restrict__ Wv2, const float* __restrict__ bv2,
    float* __restrict__ x3, float* __restrict__ val) {
    __shared__ float aT[2][16][CC + 4];
    const int t = threadIdx.x, n0 = blockIdx.x * 16;
    stage_tile<256>(h1, (size_t)n0, aT[0], t);
    stage_tile<256>(hv, (size_t)n0, aT[1], t);
    stage_fence();
    const int wave = t >> 5, lane = t & 31, l16 = lane & 15, hi = lane >> 4;
    const float* Ws[2] = {Wg2, Wv2};
    const float* Bs[2] = {bg2, bv2};
    float* Ds[2] = {x3, val};
    for (int m = 0; m < 2; m++) {
        const float* W = Ws[m];
        for (int j = 0; j < 3; j++) {
            int nb = (wave + j * 8) * 16 + l16;  // column in [0,384)
            v8f acc = {0, 0, 0, 0, 0, 0, 0, 0};
#pragma unroll 8
            for (int k0 = 0; k0 < CC; k0 += 4) {
                int ka = k0 + hi * 2;
                v2f a, b;
                a.x = aT[m][l16][ka]; a.y = aT[m][l16][ka + 1];
                b.x = W[(size_t)ka * C3 + nb];
                b.y = W[(size_t)(ka + 1) * C3 + nb];
                acc = wmma_f32(a, b, acc);
            }
            float bias = Bs[m][nb];
#pragma unroll
            for (int vv = 0; vv < 8; vv++) {
                int rr = vv + hi * 8;
                Ds[m][(size_t)(n0 + rr) * C3 + nb] = acc[vv] + bias;
            }
        }
    }
}

// ------ K3: edge GEMMs (r_attn, r_proj in LDS) + alpha + saa scatter
__global__ __launch_bounds__(256) void k_edge(
    const float* __restrict__ r, const float* __restrict__ dir,
    const float* __restrict__ dij, const int* __restrict__ ei,
    const float* __restrict__ Wre, const float* __restrict__ bre,
    const float* __restrict__ Wra, const float* __restrict__ bra,
    const float* __restrict__ q, const float* __restrict__ k,
    const float* __restrict__ x3, const float* __restrict__ vin,
    float* __restrict__ alpha, unsigned* __restrict__ mmax,
    float* __restrict__ out_s, float* __restrict__ out_v) {
    __shared__ float rT[16][CC + 4];
    __shared__ float raT[16][CC + 4];
    __shared__ float rpT[16][C3 + 8];
    __shared__ int sSrc[16], sDst[16];
    __shared__ float sCut[16];
    const int t = threadIdx.x;
    const int e0 = blockIdx.x * 16;
    stage_tile<256>(r, (size_t)e0, rT, t);
    if (t < 16) {
        int e = e0 + t;
        sSrc[t] = ei[e];
        sDst[t] = ei[EE + e];
        float d = dij[e];
        float cv = 0.5f * (__cosf(d * (3.14159265358979f / CUTOFF_R)) + 1.0f);
        sCut[t] = (d < CUTOFF_R) ? cv : 0.0f;
    }
    stage_fence();
    const int wave = t >> 5, lane = t & 31, l16 = lane & 15, hi = lane >> 4;
    {  // r_attn = silu(r @ Wre + bre), 1 col-tile per wave -> LDS
        int nb = wave * 16 + l16;
        v8f acc = {0, 0, 0, 0, 0, 0, 0, 0};
#pragma unroll 8
        for (int k0 = 0; k0 < CC; k0 += 4) {
            int ka = k0 + hi * 2;
            v2f a, b;
            a.x = rT[l16][ka]; a.y = rT[l16][ka + 1];
            b.x = Wre[(size_t)ka * CC + nb];
            b.y = Wre[(size_t)(ka + 1) * CC + nb];
            acc = wmma_f32(a, b, acc);
        }
        float bias = bre[nb];
#pragma unroll
        for (int vv = 0; vv < 8; vv++) raT[vv + hi * 8][nb] = silu_f(acc[vv] + bias);
    }
    for (int j = 0; j < 3; j++) {  // r_proj = r @ Wra + bra, 3 tiles/wave -> LDS
        int nb = (wave + j * 8) * 16 + l16;
        v8f acc = {0, 0, 0, 0, 0, 0, 0, 0};
#pragma unroll 8
        for (int k0 = 0; k0 < CC; k0 += 4) {
            int ka = k0 + hi * 2;
            v2f a, b;
            a.x = rT[l16][ka]; a.y = rT[l16][ka + 1];
            b.x = Wra[(size_t)ka * C3 + nb];
            b.y = Wra[(size_t)(ka + 1) * C3 + nb];
            acc = wmma_f32(a, b, acc);
        }
        float bias = bra[nb];
#pragma unroll
        for (int vv = 0; vv < 8; vv++) rpT[vv + hi * 8][nb] = acc[vv] + bias;
    }
    __syncthreads();
    // alpha[e,h] = sum_d q[dst]*k[src]*r_attn ; segment max via atomicMax
    if (t < 128) {
        int el = t >> 3, h = t & 7;
        int e = e0 + el;
        int sn = sSrc[el], dn = sDst[el];
        const float* qp = q + (size_t)dn * CC + h * DHH;
        const float* kp = k + (size_t)sn * CC + h * DHH;
        float sum = 0.f;
#pragma unroll
        for (int d = 0; d < DHH; d++) sum += qp[d] * kp[d] * raT[el][h * DHH + d];
        alpha[(size_t)e * HH + h] = sum;
        atomicMax(&mmax[(size_t)dn * HH + h], fenc(sum));
    }
    // saa = r_proj * x3[src] * cutoff : scatter-add into outputs (linearity)
    for (int el = 0; el < 16; el++) {
        int sn = sSrc[el], dn = sDst[el];
        int e = e0 + el;
        float cut = sCut[el];
        for (int c = t; c < C3; c += 256) {
            float m = rpT[el][c] * x3[(size_t)sn * C3 + c] * cut;
            if (c < CC) {
                atomicAdd(&out_s[(size_t)dn * CC + c], m);
            } else if (c < 2 * CC) {
                int cd = c - CC;
#pragma unroll
                for (int a = 0; a < 3; a++)
                    atomicAdd(&out_v[((size_t)dn * 3 + a) * CC + cd],
                              dir[(size_t)e * 3 + a] * m);
            } else {
                int cv = c - 2 * CC;
#pragma unroll
                for (int a = 0; a < 3; a++)
                    atomicAdd(&out_v[((size_t)dn * 3 + a) * CC + cv],
                              m * vin[((size_t)sn * 3 + a) * CC + cv]);
            }
        }
    }
}

// ------------------------------------------------ K4: exp + segment z-sum
__global__ __launch_bounds__(256) void k_exp(const int* __restrict__ ei,
                                             float* __restrict__ alpha,
                                             const unsigned* __restrict__ mmax,
                                             float* __restrict__ zsum) {
    int i = blockIdx.x * 256 + threadIdx.x;
    if (i >= EE * HH) return;
    int e = i >> 3, h = i & 7;
    int dn = ei[EE + e];
    float m = fdec(mmax[(size_t)dn * HH + h]);
    if (m < -3.0e38f) m = 0.0f;
    float ex = __expf(alpha[i] - m);
    alpha[i] = ex;
    atomicAdd(&zsum[(size_t)dn * HH + h], ex);
}

// --------------------------------------------------------- K5: normalize
__global__ __launch_bounds__(256) void k_norm(const int* __restrict__ ei,
                                              float* __restrict__ alpha,
                                              const float* __restrict__ zsum) {
    int i = blockIdx.x * 256 + threadIdx.x;
    if (i >= EE * HH) return;
    int e = i >> 3, h = i & 7;
    int dn = ei[EE + e];
    alpha[i] = alpha[i] / (zsum[(size_t)dn * HH + h] + 1e-16f);
}

// ------------------------------------------- K6: sea = attn*val[src] scatter
__global__ __launch_bounds__(256) void k_sea(
    const int* __restrict__ ei, const float* __restrict__ dir,
    const float* __restrict__ attn, const float* __restrict__ val,
    const float* __restrict__ vin, float* __restrict__ out_s,
    float* __restrict__ out_v) {
    __shared__ int sSrc[16], sDst[16];
    __shared__ float sAt[16][8];
    const int t = threadIdx.x;
    const int e0 = blockIdx.x * 16;
    if (t < 16) { sSrc[t] = ei[e0 + t]; sDst[t] = ei[EE + e0 + t]; }
    if (t < 128) {
        int el = t >> 3, h = t & 7;
        sAt[el][h] = attn[(size_t)(e0 + el) * HH + h];
    }
    __syncthreads();
    for (int el = 0; el < 16; el++) {
        int sn = sSrc[el], dn = sDst[el];
        int e = e0 + el;
        for (int c = t; c < C3; c += 256) {
            float m = sAt[el][c / 48] * val[(size_t)sn * C3 + c];
            if (c < CC) {
                atomicAdd(&out_s[(size_t)dn * CC + c], m);
            } else if (c < 2 * CC) {
                int cd = c - CC;
#pragma unroll
                for (int a = 0; a < 3; a++)
                    atomicAdd(&out_v[((size_t)dn * 3 + a) * CC + cd],
                              dir[(size_t)e * 3 + a] * m);
            } else {
                int cv = c - 2 * CC;
#pragma unroll
                for (int a = 0; a < 3; a++)
                    atomicAdd(&out_v[((size_t)dn * 3 + a) * CC + cv],
                              m * vin[((size_t)sn * 3 + a) * CC + cv]);
            }
        }
    }
}

extern "C" void kernel_launch(void* const* d_in, const int* in_sizes, int n_in,
                              void* d_out, int out_size, void* d_ws,
                              size_t ws_size, hipStream_t stream) {
    const float* s    = (const float*)d_in[0];
    const float* vin  = (const float*)d_in[1];
    const float* dir  = (const float*)d_in[2];
    const float* dij  = (const float*)d_in[3];
    const float* rij  = (const float*)d_in[4];
    const int*   ei   = (const int*)d_in[5];
    const float* ln_g = (const float*)d_in[6];
    const float* ln_b = (const float*)d_in[7];
    const float* Wq = (const float*)d_in[8],  *bq = (const float*)d_in[9];
    const float* Wk = (const float*)d_in[10], *bk = (const float*)d_in[11];
    const float* Wg1 = (const float*)d_in[12], *bg1 = (const float*)d_in[13];
    const float* Wg2 = (const float*)d_in[14], *bg2 = (const float*)d_in[15];
    const float* Wv1 = (const float*)d_in[16], *bv1 = (const float*)d_in[17];
    const float* Wv2 = (const float*)d_in[18], *bv2 = (const float*)d_in[19];
    const float* Wre = (const float*)d_in[20], *bre = (const float*)d_in[21];
    const float* Wra = (const float*)d_in[22], *bra = (const float*)d_in[23];

    float* out_s = (float*)d_out;               // N*C
    float* out_v = out_s + (size_t)NN * CC;     // N*3*C
    float* out_r = out_v + (size_t)NN * 3 * CC; // E*C

    // workspace layout (floats)
    float* ws = (float*)d_ws;
    const size_t NC = (size_t)NN * CC;
    float* q     = ws;
    float* k     = q + NC;
    float* h1    = k + NC;
    float* hv    = h1 + NC;
    float* x3    = hv + NC;            // N*3C
    float* val   = x3 + (size_t)NN * C3;
    float* alpha = val + (size_t)NN * C3;              // E*H
    unsigned* mmax = (unsigned*)(alpha + (size_t)EE * HH);  // N*H
    float* zsum  = (float*)(mmax + (size_t)NN * HH);        // N*H

    k_init<<<(EE * CC / 4 + 255) / 256, 256, 0, stream>>>(rij, vin, out_v, out_r,
                                                          mmax, zsum);
    k_node1<<<NN / 16, 256, 0, stream>>>(s, ln_g, ln_b, Wq, bq, Wk, bk, Wg1, bg1,
                                         Wv1, bv1, out_s, q, k, h1, hv);
    k_node2<<<NN / 16, 256, 0, stream>>>(h1, hv, Wg2, bg2, Wv2, bv2, x3, val);
    k_edge<<<EE / 16, 256, 0, stream>>>(rij, dir, dij, ei, Wre, bre, Wra, bra, q,
                                        k, x3, vin, alpha, mmax, out_s, out_v);
    k_exp<<<(EE * HH + 255) / 256, 256, 0, stream>>>(ei, alpha, mmax, zsum);
    k_norm<<<(EE * HH + 255) / 256, 256, 0, stream>>>(ei, alpha, zsum);
    k_sea<<<EE / 16, 256, 0, stream>>>(ei, dir, alpha, val, vin, out_s, out_v);
}